// RGate_549755814191
// MI455X (gfx1250) — compile-verified
//
#include <hip/hip_runtime.h>
#include <math.h>

// ---------------------------------------------------------------------------
// RGate on 22 qubits, state = 2^22 complex64 stored as interleaved float pairs.
//
// U = kron_i expm(-0.5j * theta_i * X).  Fuse 3 qubits per pass:
//   W(8x8 complex) = M_b2 (x) M_b1 (x) M_b0
//   real form A(16x16) = [[Re W, -Im W],[Im W, Re W]] acting on [re(0..7); im(0..7)]
// Each pass: one 16x16 real matmul per 16 columns via 4x V_WMMA_F32_16X16X4_F32.
// 8 in-place passes total (7 groups of 3 qubits + M21 (x) I (x) I).
// State (33.5 MB) is L2-resident (192 MB), so passes run at L2 bandwidth.
// ---------------------------------------------------------------------------

typedef __attribute__((ext_vector_type(2))) float v2f;
typedef __attribute__((ext_vector_type(8))) float v8f;

#define QL 22
#define NCOLS  (1u << (QL - 3))    // 2^19 columns per pass
#define NTILES (NCOLS / 16u)       // 32768 16-column tiles per pass
#define PASS_BLOCKS 512u           // 512 blocks * 8 waves -> 8 tiles per wave

// Build per-group 16x16 real matrices into mats[g*256 + row*16 + col].
// Group g<7: bits (3g,3g+1,3g+2) all active.  Group 7: bits (19,20,21), only
// bit 21 active (identity on 19,20 -> harmless re-touch).
// Reference mapping: flat-index bit k is rotated with angle[21-k].
__global__ void rgate_setup(const float* __restrict__ angle,
                            float* __restrict__ mats) {
  const int g   = blockIdx.x;
  const int j   = threadIdx.x;
  const int row = j >> 4, col = j & 15;
  const int t = row & 7, tp = col & 7;
  const int base = (g < 7) ? 3 * g : 19;

  float wr = 1.0f, wi = 0.0f;        // W[t][tp] accumulated as complex product
  for (int k = 0; k < 3; ++k) {
    const int  bit    = base + k;
    const bool active = (g < 7) || (k == 2);
    const int tk = (t >> k) & 1, tpk = (tp >> k) & 1;
    float mr, mi;
    if (!active) {
      mr = (tk == tpk) ? 1.0f : 0.0f; mi = 0.0f;
    } else {
      const float th = 0.5f * angle[(QL - 1) - bit];
      float s, c;
      __sincosf(th, &s, &c);
      if (tk == tpk) { mr = c;    mi = 0.0f; }
      else           { mr = 0.0f; mi = -s;   }
    }
    const float nr = wr * mr - wi * mi;
    const float ni = wr * mi + wi * mr;
    wr = nr; wi = ni;
  }
  const int qr = row >> 3, qc = col >> 3;
  const float val = (qr == 0) ? ((qc == 0) ? wr : -wi)
                              : ((qc == 0) ? wi :  wr);
  mats[g * 256 + row * 16 + col] = val;
}

// One fused-3-qubit pass.  Each wave32 grid-strides over 16-column tiles;
// per column the 8 complex amplitudes live at indices formed by inserting the
// 3 group bits into the column index r.  In-place safe: each tile's indices
// are owned by exactly one wave and all its loads precede its stores.
__global__ __launch_bounds__(256) void rgate_pass(
    const float* __restrict__ in, float* __restrict__ out,
    const float* __restrict__ Am, int b0, int b1, int b2) {
  const int tid  = threadIdx.x;
  const int wave = tid >> 5;
  const int lane = tid & 31;
  const int m    = lane & 15;   // column within tile / matrix row
  const int half = lane >> 4;   // 0: K0/K1 + real-part rows, 1: K2/K3 + imag rows

  // A operand, loaded once and reused for all tiles this wave processes:
  // 16x4 tile per WMMA; lane row m, cols 4t + 2*half + {0,1}.
  const float* arow = Am + m * 16 + 2 * half;
  v2f a0, a1, a2, a3;
  a0.x = arow[0];  a0.y = arow[1];
  a1.x = arow[4];  a1.y = arow[5];
  a2.x = arow[8];  a2.y = arow[9];
  a3.x = arow[12]; a3.y = arow[13];

  const float2* in2 = (const float2*)in;
  const unsigned waveSlots = PASS_BLOCKS * 8u;   // total wave slots in grid

  for (unsigned tile = blockIdx.x * 8u + (unsigned)wave; tile < NTILES;
       tile += waveSlots) {
    const unsigned r = tile * 16u + (unsigned)m;

    // Insert zero bits at b0 < b1 < b2 (final-position ascending insertion).
    unsigned e = r;
    e = ((e >> b0) << (b0 + 1)) | (e & ((1u << b0) - 1u));
    e = ((e >> b1) << (b1 + 1)) | (e & ((1u << b1) - 1u));
    e = ((e >> b2) << (b2 + 1)) | (e & ((1u << b2) - 1u));

    auto idx_of = [&](int t) -> unsigned {
      return e | ((unsigned)(t & 1) << b0)
               | ((unsigned)((t >> 1) & 1) << b1)
               | ((unsigned)((t >> 2) & 1) << b2);
    };

    // B operand: lane half 0 loads amplitudes t={0,1,4,5}, half 1 -> {2,3,6,7}.
    const int tb = half * 2;
    const float2 c0 = in2[idx_of(tb + 0)];
    const float2 c1 = in2[idx_of(tb + 1)];
    const float2 c2 = in2[idx_of(tb + 4)];
    const float2 c3 = in2[idx_of(tb + 5)];
    v2f bv0, bv1, bv2, bv3;
    bv0.x = c0.x; bv0.y = c1.x;   // components 0..3:   re(a0..a3)
    bv1.x = c2.x; bv1.y = c3.x;   // components 4..7:   re(a4..a7)
    bv2.x = c0.y; bv2.y = c1.y;   // components 8..11:  im(a0..a3)
    bv3.x = c2.y; bv3.y = c3.y;   // components 12..15: im(a4..a7)

    // D(16x16) = A(16x16) * B(16x16) via 4 accumulating K=4 f32 WMMAs.
    v8f acc = {};
    acc = __builtin_amdgcn_wmma_f32_16x16x4_f32(false, a0, false, bv0, (short)0, acc, false, false);
    acc = __builtin_amdgcn_wmma_f32_16x16x4_f32(false, a1, false, bv1, (short)0, acc, false, false);
    acc = __builtin_amdgcn_wmma_f32_16x16x4_f32(false, a2, false, bv2, (short)0, acc, false, false);
    acc = __builtin_amdgcn_wmma_f32_16x16x4_f32(false, a3, false, bv3, (short)0, acc, false, false);

    // D row (v + 8*half), col m:  half==0 -> re(out a_v), half==1 -> im(out a_v).
#pragma unroll
    for (int v = 0; v < 8; ++v) {
      const unsigned iv = idx_of(v);
      out[2u * iv + (unsigned)half] = acc[v];
    }
  }
}

extern "C" void kernel_launch(void* const* d_in, const int* in_sizes, int n_in,
                              void* d_out, int out_size, void* d_ws, size_t ws_size,
                              hipStream_t stream) {
  const float* x     = (const float*)d_in[0];   // [2^22][2] float32
  const float* angle = (const float*)d_in[1];   // [22] float32
  float* out  = (float*)d_out;                  // [2^22][2] float32
  float* mats = (float*)d_ws;                   // 8 * 256 floats = 8 KB

  rgate_setup<<<dim3(8), dim3(256), 0, stream>>>(angle, mats);

  for (int g = 0; g < 8; ++g) {
    const int base = (g < 7) ? 3 * g : 19;
    const float* src = (g == 0) ? x : out;      // pass 0 reads input, rest in-place
    rgate_pass<<<dim3(PASS_BLOCKS), dim3(256), 0, stream>>>(
        src, out, mats + g * 256, base, base + 1, base + 2);
  }
}